// PreprocessAndCalculateModel_61933428412402
// MI455X (gfx1250) — compile-verified
//
#include <hip/hip_runtime.h>
#include <math.h>

#define BB 8
#define NN 2048
#define NITER_ 40

typedef __attribute__((ext_vector_type(2))) float v2f;
typedef __attribute__((ext_vector_type(8))) float v8f;

// ---------------------------------------------------------------------------
// K1: W = A @ v per batch, driven by V_WMMA_F32_16X16X4_F32.
// Each wave owns a 16-row tile and a K-slice of 256. B operand broadcasts
// v[k] across all 16 columns, so every column of D equals the partial matvec.
// fp32 A-operand layout (ISA 7.12.2): VGPR0 = K0 (lanes 0-15) / K2 (16-31),
// VGPR1 = K1 / K3 -> one b64 load per lane lands directly in layout.
// C/D layout: lanes 0-15 hold rows M=0..7 in VGPRs 0..7, lanes 16-31 M=8..15.
// ---------------------------------------------------------------------------
#define MV_WAVES 8
#define MV_THREADS 256
#define KSLICE (NN / MV_WAVES) // 256

__global__ __launch_bounds__(MV_THREADS)
void lobpcg_matvec_wmma(const float* __restrict__ A,
                        const float* __restrict__ v,
                        float* __restrict__ w) {
  const int batch = blockIdx.y;
  const int row0  = blockIdx.x << 4;
  const int wave  = threadIdx.x >> 5;
  const int lane  = threadIdx.x & 31;
  const int half  = lane >> 4;      // 0: K%4 in {0,1}, 1: K%4 in {2,3}
  const int r     = lane & 15;      // row within 16-row tile

  const float* __restrict__ Ab   = A + (size_t)batch * NN * NN;
  const float* __restrict__ vb   = v + (size_t)batch * NN;
  const float* __restrict__ arow = Ab + (size_t)(row0 + r) * NN;
  const int koff  = half << 1;
  const int kbase = wave * KSLICE;

  v8f acc0 = {}, acc1 = {}, acc2 = {}, acc3 = {};
  for (int k = kbase; k < kbase + KSLICE; k += 16) {
    const v2f a0 = *(const v2f*)(arow + k + 0  + koff);
    const v2f b0 = *(const v2f*)(vb   + k + 0  + koff);
    const v2f a1 = *(const v2f*)(arow + k + 4  + koff);
    const v2f b1 = *(const v2f*)(vb   + k + 4  + koff);
    const v2f a2 = *(const v2f*)(arow + k + 8  + koff);
    const v2f b2 = *(const v2f*)(vb   + k + 8  + koff);
    const v2f a3 = *(const v2f*)(arow + k + 12 + koff);
    const v2f b3 = *(const v2f*)(vb   + k + 12 + koff);
    acc0 = __builtin_amdgcn_wmma_f32_16x16x4_f32(false, a0, false, b0, (short)0, acc0, false, false);
    acc1 = __builtin_amdgcn_wmma_f32_16x16x4_f32(false, a1, false, b1, (short)0, acc1, false, false);
    acc2 = __builtin_amdgcn_wmma_f32_16x16x4_f32(false, a2, false, b2, (short)0, acc2, false, false);
    acc3 = __builtin_amdgcn_wmma_f32_16x16x4_f32(false, a3, false, b3, (short)0, acc3, false, false);
  }
  v8f acc = (acc0 + acc1) + (acc2 + acc3);

  // Cross-wave K reduction through LDS. All D columns identical, so lane 0
  // (rows 0..7) and lane 16 (rows 8..15) carry complete per-wave row sums.
  __shared__ float red[MV_WAVES][16];
  if ((lane & 15) == 0) {
#pragma unroll
    for (int i = 0; i < 8; ++i) red[wave][(half << 3) + i] = acc[i];
  }
  __syncthreads();
  if (threadIdx.x < 16) {
    float s = 0.f;
#pragma unroll
    for (int wv = 0; wv < MV_WAVES; ++wv) s += red[wv][threadIdx.x];
    w[(size_t)batch * NN + row0 + threadIdx.x] = s;
  }
}

// ---------------------------------------------------------------------------
// K2: Gram sums for S=[X,AX,P], AS=[AX,W,AP]. 12 unique length-N dots/batch.
// ---------------------------------------------------------------------------
__global__ __launch_bounds__(256)
void lobpcg_gram(const float* __restrict__ X, const float* __restrict__ AX,
                 const float* __restrict__ P, const float* __restrict__ AP,
                 const float* __restrict__ W, float* __restrict__ sums) {
  const int b = blockIdx.x;
  const int t = threadIdx.x;
  const size_t base = (size_t)b * NN;
  float s[12];
#pragma unroll
  for (int j = 0; j < 12; ++j) s[j] = 0.f;
  for (int i = t; i < NN; i += 256) {
    const float xi = X[base + i], axi = AX[base + i], pi = P[base + i];
    const float api = AP[base + i], wi = W[base + i];
    s[0]  += xi * xi;   s[1]  += xi * axi;  s[2]  += xi * pi;
    s[3]  += axi * axi; s[4]  += axi * pi;  s[5]  += pi * pi;
    s[6]  += xi * wi;   s[7]  += xi * api;  s[8]  += axi * wi;
    s[9]  += axi * api; s[10] += pi * wi;   s[11] += pi * api;
  }
  __shared__ float lds[256];
#pragma unroll
  for (int j = 0; j < 12; ++j) {
    lds[t] = s[j];
    __syncthreads();
    for (int off = 128; off > 0; off >>= 1) {
      if (t < off) lds[t] += lds[t + off];
      __syncthreads();
    }
    if (t == 0) sums[b * 12 + j] = lds[0];
    __syncthreads();
  }
}

// ---------------------------------------------------------------------------
// K3: per-batch 3x3 Rayleigh-Ritz. Cholesky-whiten G, symmetric eig via
// shifted power iteration, emit combination coefficients y (new X/AX) and
// q (new P/AP), plus the Ritz value.
// ---------------------------------------------------------------------------
__global__ void lobpcg_solve(const float* __restrict__ sums,
                             float* __restrict__ coef,
                             float* __restrict__ Eout, int last) {
  const int b = threadIdx.x;
  if (b >= BB) return;
  const float* s = sums + b * 12;
  float G00 = s[0], G01 = s[1], G02 = s[2], G11 = s[3], G12 = s[4], G22 = s[5];
  // H = S^T A S (symmetrized using exact-arithmetic equalities)
  const float H00 = s[1];
  const float H01 = 0.5f * (s[6] + s[3]);
  const float H02 = 0.5f * (s[7] + s[4]);
  const float H11 = s[8];
  const float H12 = 0.5f * (s[9] + s[10]);
  const float H22 = s[11];

  const float eps = 1e-10f * (G00 + G11 + G22) + 1e-30f;
  G00 += eps; G11 += eps; G22 += eps;
  // Cholesky G = L L^T
  const float L00 = sqrtf(fmaxf(G00, 1e-30f));
  const float L10 = G01 / L00;
  const float L20 = G02 / L00;
  const float L11 = sqrtf(fmaxf(G11 - L10 * L10, eps));
  const float L21 = (G12 - L20 * L10) / L11;
  const float L22 = sqrtf(fmaxf(G22 - L20 * L20 - L21 * L21, eps));
  // M = inv(L) (lower), U = M^T (upper) so that Q = S U is orthonormal
  const float M00 = 1.f / L00, M11 = 1.f / L11, M22 = 1.f / L22;
  const float M10 = -L10 * M00 * M11;
  const float M21 = -L21 * M11 * M22;
  const float M20 = (L10 * L21 - L20 * L11) * M00 * M11 * M22;
  float U[3][3] = {{M00, M10, M20}, {0.f, M11, M21}, {0.f, 0.f, M22}};
  float Hm[3][3] = {{H00, H01, H02}, {H01, H11, H12}, {H02, H12, H22}};
  float HU[3][3], T[3][3];
#pragma unroll
  for (int i = 0; i < 3; ++i)
#pragma unroll
    for (int j = 0; j < 3; ++j)
      HU[i][j] = Hm[i][0] * U[0][j] + Hm[i][1] * U[1][j] + Hm[i][2] * U[2][j];
#pragma unroll
  for (int i = 0; i < 3; ++i)
#pragma unroll
    for (int j = 0; j < 3; ++j)
      T[i][j] = U[0][i] * HU[0][j] + U[1][i] * HU[1][j] + U[2][i] * HU[2][j];
  const float T00 = T[0][0], T11 = T[1][1], T22 = T[2][2];
  const float T01 = 0.5f * (T[0][1] + T[1][0]);
  const float T02 = 0.5f * (T[0][2] + T[2][0]);
  const float T12 = 0.5f * (T[1][2] + T[2][1]);
  // Gershgorin shift makes T + sig*I PSD -> power iteration finds lambda_max
  float sig = fabsf(T00) + fabsf(T01) + fabsf(T02);
  sig = fmaxf(sig, fabsf(T01) + fabsf(T11) + fabsf(T12));
  sig = fmaxf(sig, fabsf(T02) + fabsf(T12) + fabsf(T22));
  float c0 = 0.57735027f, c1 = 0.57735027f, c2 = 0.57735027f;
  for (int it = 0; it < 64; ++it) {
    const float d0 = (T00 + sig) * c0 + T01 * c1 + T02 * c2;
    const float d1 = T01 * c0 + (T11 + sig) * c1 + T12 * c2;
    const float d2 = T02 * c0 + T12 * c1 + (T22 + sig) * c2;
    const float inv = 1.f / sqrtf(fmaxf(d0 * d0 + d1 * d1 + d2 * d2, 1e-30f));
    c0 = d0 * inv; c1 = d1 * inv; c2 = d2 * inv;
  }
  const float lam = c0 * (T00 * c0 + T01 * c1 + T02 * c2) +
                    c1 * (T01 * c0 + T11 * c1 + T12 * c2) +
                    c2 * (T02 * c0 + T12 * c1 + T22 * c2);
  // y: coefficients of new X in [X,AX,P] basis; normalize ||S y|| = 1
  float y0 = U[0][0] * c0 + U[0][1] * c1 + U[0][2] * c2;
  float y1 = U[1][1] * c1 + U[1][2] * c2;
  float y2 = U[2][2] * c2;
  float gy0 = G00 * y0 + G01 * y1 + G02 * y2;
  float gy1 = G01 * y0 + G11 * y1 + G12 * y2;
  float gy2 = G02 * y0 + G12 * y1 + G22 * y2;
  const float sX = 1.f / fmaxf(sqrtf(fmaxf(y0 * gy0 + y1 * gy1 + y2 * gy2, 0.f)), 1e-8f);
  y0 *= sX; y1 *= sX; y2 *= sX;
  // Pn = Xn - Xhat (Xhat^T Xn): q = y - e0 * (g0^T y)/G00, then normalize
  const float alpha = (G00 * y0 + G01 * y1 + G02 * y2) / G00;
  float q0 = y0 - alpha, q1 = y1, q2 = y2;
  gy0 = G00 * q0 + G01 * q1 + G02 * q2;
  gy1 = G01 * q0 + G11 * q1 + G12 * q2;
  gy2 = G02 * q0 + G12 * q1 + G22 * q2;
  const float sP = 1.f / fmaxf(sqrtf(fmaxf(q0 * gy0 + q1 * gy1 + q2 * gy2, 0.f)), 1e-8f);
  q0 *= sP; q1 *= sP; q2 *= sP;

  float* c = coef + b * 8;
  c[0] = y0; c[1] = y1; c[2] = y2;
  c[3] = q0; c[4] = q1; c[5] = q2;
  c[6] = lam;
  if (last) Eout[b] = lam;
}

// ---------------------------------------------------------------------------
// K4: rank-3 recombination: new (X,AX,P,AP) from old basis + W, ping-pong.
// ---------------------------------------------------------------------------
__global__ __launch_bounds__(256)
void lobpcg_update(const float* __restrict__ Xc, const float* __restrict__ AXc,
                   const float* __restrict__ Pc, const float* __restrict__ APc,
                   const float* __restrict__ W, const float* __restrict__ coef,
                   float* __restrict__ Xn, float* __restrict__ AXn,
                   float* __restrict__ Pn, float* __restrict__ APn,
                   float* __restrict__ Xout, int last) {
  const int b = blockIdx.y;
  const int i = blockIdx.x * 256 + threadIdx.x;
  const float* c = coef + b * 8;
  const float y0 = c[0], y1 = c[1], y2 = c[2];
  const float q0 = c[3], q1 = c[4], q2 = c[5];
  const size_t idx = (size_t)b * NN + i;
  const float x = Xc[idx], ax = AXc[idx], p = Pc[idx], ap = APc[idx], w = W[idx];
  const float xn  = x * y0 + ax * y1 + p * y2;
  const float axn = ax * y0 + w * y1 + ap * y2;
  const float pn  = x * q0 + ax * q1 + p * q2;
  const float apn = ax * q0 + w * q1 + ap * q2;
  Xn[idx] = xn; AXn[idx] = axn; Pn[idx] = pn; APn[idx] = apn;
  if (last) Xout[idx] = xn;
}

// ---------------------------------------------------------------------------
// Init: X = x0/||x0||, P = p0/||p0||
// ---------------------------------------------------------------------------
__global__ __launch_bounds__(256)
void lobpcg_init(const float* __restrict__ x0, const float* __restrict__ p0,
                 float* __restrict__ X, float* __restrict__ P) {
  const int b = blockIdx.x;
  const int t = threadIdx.x;
  const size_t base = (size_t)b * NN;
  float sx = 0.f, sp = 0.f;
  for (int i = t; i < NN; i += 256) {
    const float a = x0[base + i]; sx += a * a;
    const float c = p0[base + i]; sp += c * c;
  }
  __shared__ float lx[256], lp[256];
  lx[t] = sx; lp[t] = sp;
  __syncthreads();
  for (int off = 128; off > 0; off >>= 1) {
    if (t < off) { lx[t] += lx[t + off]; lp[t] += lp[t + off]; }
    __syncthreads();
  }
  const float ix = 1.f / fmaxf(sqrtf(lx[0]), 1e-8f);
  const float ip = 1.f / fmaxf(sqrtf(lp[0]), 1e-8f);
  for (int i = t; i < NN; i += 256) {
    X[base + i] = x0[base + i] * ix;
    P[base + i] = p0[base + i] * ip;
  }
}

extern "C" void kernel_launch(void* const* d_in, const int* in_sizes, int n_in,
                              void* d_out, int out_size, void* d_ws, size_t ws_size,
                              hipStream_t stream) {
  const float* A  = (const float*)d_in[0]; // (B, N, N)
  const float* x0 = (const float*)d_in[1]; // (B, N, 1)
  const float* p0 = (const float*)d_in[2]; // (B, N, 1)
  float* out  = (float*)d_out;
  float* E    = out;        // (B, 1)
  float* Xout = out + BB;   // (B, N, 1)

  float* ws = (float*)d_ws;
  const size_t VB = (size_t)BB * NN;
  // ping-pong basis buffers: [X | AX | P | AP] each VB floats
  float* buf0 = ws;
  float* buf1 = ws + 4 * VB;
  float* W    = ws + 8 * VB;       // A @ AX scratch
  float* sums = W + VB;            // 12 per batch
  float* coef = sums + BB * 12;    // 8 per batch

  const dim3 mvGrid(NN / 16, BB);
  const dim3 mvBlk(MV_THREADS);

  lobpcg_init<<<BB, 256, 0, stream>>>(x0, p0, buf0, buf0 + 2 * VB);
  lobpcg_matvec_wmma<<<mvGrid, mvBlk, 0, stream>>>(A, buf0, buf0 + VB);              // AX
  lobpcg_matvec_wmma<<<mvGrid, mvBlk, 0, stream>>>(A, buf0 + 2 * VB, buf0 + 3 * VB); // AP

  int cur = 0;
  for (int it = 0; it < NITER_; ++it) {
    const int last = (it == NITER_ - 1) ? 1 : 0;
    float* C = cur ? buf1 : buf0;
    float* Nw = cur ? buf0 : buf1;
    lobpcg_matvec_wmma<<<mvGrid, mvBlk, 0, stream>>>(A, C + VB, W); // W = A @ AX
    lobpcg_gram<<<BB, 256, 0, stream>>>(C, C + VB, C + 2 * VB, C + 3 * VB, W, sums);
    lobpcg_solve<<<1, 32, 0, stream>>>(sums, coef, E, last);
    lobpcg_update<<<dim3(NN / 256, BB), 256, 0, stream>>>(
        C, C + VB, C + 2 * VB, C + 3 * VB, W, coef,
        Nw, Nw + VB, Nw + 2 * VB, Nw + 3 * VB, Xout, last);
    cur ^= 1;
  }
}